// attention_net_46471546142993
// MI455X (gfx1250) — compile-verified
//
#include <hip/hip_runtime.h>
#include <hip/hip_bf16.h>

#define N 8192
#define FIN 128
#define FOUT 128
#define ALPHA 0.2f
#define NEG_INF -9000000000000000.0f

typedef float v2f __attribute__((ext_vector_type(2)));
typedef float v8f __attribute__((ext_vector_type(8)));

// ---------------------------------------------------------------------------
// Kernel 1: fold the attention vectors into W.
//   wa1[f] = sum_o W[f][o] * a[o]        (o in [0,128))
//   wa2[f] = sum_o W[f][o] * a[128+o]
// One block of 256 threads; trivial cost.
// ---------------------------------------------------------------------------
__global__ void make_wa_kernel(const float* __restrict__ W,
                               const float* __restrict__ a,
                               float* __restrict__ wa1,
                               float* __restrict__ wa2) {
    int t = threadIdx.x;          // 0..255
    int f = t & (FIN - 1);
    const float* av = (t < FIN) ? a : (a + FOUT);
    float acc = 0.f;
#pragma unroll 8
    for (int o = 0; o < FOUT; ++o)
        acc = fmaf(W[f * FOUT + o], av[o], acc);
    if (t < FIN) wa1[f] = acc;
    else         wa2[f] = acc;
}

// ---------------------------------------------------------------------------
// Kernel 2: s[row] = dot(V[row][0:128], wa[0:128]) via V_WMMA_F32_16X16X4_F32.
// One wave (32 lanes) per 16-row tile. A fragment = 16x4 fp32 tile of V
// (VGPR0: lanes0-15 K=kb+0, lanes16-31 K=kb+2; VGPR1: K=kb+1 / K=kb+3).
// B fragment = 4x16 with column 0 = wa (same K striping). Column 0 of the
// 16x16 D accumulator (lanes 0 and 16, VGPRs 0..7) holds the 16 dot products.
// blockIdx.y: 0 -> (v_i, wa1, s_i), 1 -> (v_j, wa2, s_j).
// ---------------------------------------------------------------------------
__global__ void rowdot_wmma_kernel(const float* __restrict__ v_i,
                                   const float* __restrict__ v_j,
                                   const float* __restrict__ wa1,
                                   const float* __restrict__ wa2,
                                   float* __restrict__ s_i,
                                   float* __restrict__ s_j) {
    const float* V  = (blockIdx.y == 0) ? v_i : v_j;
    const float* wa = (blockIdx.y == 0) ? wa1 : wa2;
    float*       s  = (blockIdx.y == 0) ? s_i : s_j;

    const int lane = threadIdx.x;        // 0..31, all lanes active (EXEC all-1)
    const int m    = lane & 15;          // row within tile
    const int kh   = lane >> 4;          // K-half select (0 -> K+0/1, 1 -> K+2/3)

    const float* vrow = V + (size_t)(blockIdx.x * 16 + m) * FIN;

    v8f c = {};
#pragma unroll
    for (int kb = 0; kb < FIN; kb += 4) {
        const int k0 = kb + 2 * kh;
        v2f av;
        av.x = vrow[k0];
        av.y = vrow[k0 + 1];
        const float b0 = wa[k0];
        const float b1 = wa[k0 + 1];
        v2f bv;
        bv.x = (m == 0) ? b0 : 0.f;      // B column 0 = wa, other columns 0
        bv.y = (m == 0) ? b1 : 0.f;
        // D = A*B + C  (fp32 WMMA, 8-arg pattern: neg_a, A, neg_b, B, c_mod, C, reuse_a, reuse_b)
        c = __builtin_amdgcn_wmma_f32_16x16x4_f32(false, av, false, bv,
                                                  (short)0, c, false, false);
    }

    // Column N=0 lives in lane 0 (M = r) and lane 16 (M = 8 + r), VGPR r.
    if (m == 0) {
        const int base = blockIdx.x * 16 + kh * 8;
#pragma unroll
        for (int r = 0; r < 8; ++r)
            s[base + r] = c[r];
    }
}

// ---------------------------------------------------------------------------
// Kernel 3: the bandwidth-bound pass. One block (256 threads) per row i:
//   score[j] = adj[i][j] > 0 ? leaky_relu(s_i[i] + s_j[j]) : -9e15
//   out[i][j] = softmax_j(score)
// adj row is read once with NT loads; out written once with NT stores.
// s_j (32 KB) is staged in LDS. Per-thread register tile of 32 values.
// ---------------------------------------------------------------------------
#define TPB 256
#define VPT (N / TPB)   // 32 values per thread

__device__ __forceinline__ float wave_max(float v) {
#pragma unroll
    for (int m = 16; m > 0; m >>= 1)
        v = fmaxf(v, __shfl_xor(v, m, 32));
    return v;
}
__device__ __forceinline__ float wave_sum(float v) {
#pragma unroll
    for (int m = 16; m > 0; m >>= 1)
        v += __shfl_xor(v, m, 32);
    return v;
}

__global__ void __launch_bounds__(TPB)
gat_softmax_kernel(const int* __restrict__ adj,
                   const float* __restrict__ s_i,
                   const float* __restrict__ s_j,
                   float* __restrict__ out) {
    __shared__ float sj[N];       // 32 KB
    __shared__ float red[TPB / 32];

    const int tid = threadIdx.x;
    const int row = blockIdx.x;

    for (int j = tid; j < N; j += TPB)
        sj[j] = s_j[j];
    __syncthreads();

    const float si = s_i[row];
    const int*   arow = adj + (size_t)row * N;
    float*       orow = out + (size_t)row * N;

    float vals[VPT];
    float lmax = NEG_INF;
#pragma unroll
    for (int k = 0; k < VPT; ++k) {
        const int j = tid + k * TPB;                 // coalesced, stride TPB
        const int av = __builtin_nontemporal_load(arow + j);
        const float x = si + sj[j];
        const float e = (x >= 0.f) ? x : ALPHA * x;  // leaky_relu
        const float v = (av > 0) ? e : NEG_INF;
        vals[k] = v;
        lmax = fmaxf(lmax, v);
    }

    // block-wide max
    float wmax = wave_max(lmax);
    const int wid  = tid >> 5;
    const int lane = tid & 31;
    if (lane == 0) red[wid] = wmax;
    __syncthreads();
    float rmax = red[0];
#pragma unroll
    for (int w = 1; w < TPB / 32; ++w) rmax = fmaxf(rmax, red[w]);
    __syncthreads();

    // exp + block-wide sum (exp(NEG_INF - rmax) underflows to 0 when rmax finite;
    // when the row has no neighbors rmax == NEG_INF and all terms become 1 ->
    // uniform 1/N, matching jax softmax semantics)
    float lsum = 0.f;
#pragma unroll
    for (int k = 0; k < VPT; ++k) {
        const float ev = __expf(vals[k] - rmax);
        vals[k] = ev;
        lsum += ev;
    }
    float wsum = wave_sum(lsum);
    if (lane == 0) red[wid] = wsum;
    __syncthreads();
    float rsum = 0.f;
#pragma unroll
    for (int w = 0; w < TPB / 32; ++w) rsum += red[w];

    const float inv = 1.f / rsum;
#pragma unroll
    for (int k = 0; k < VPT; ++k) {
        const int j = tid + k * TPB;
        __builtin_nontemporal_store(vals[k] * inv, orow + j);
    }
}

// ---------------------------------------------------------------------------
// Host-side launcher. Inputs: v_i (8192x128 f32), v_j (8192x128 f32),
// adj (8192x8192 i32), W (128x128 f32), a (256x1 f32). Output: 8192x8192 f32.
// Workspace: wa1[128] | wa2[128] | s_i[8192] | s_j[8192]  (66.5 KB)
// ---------------------------------------------------------------------------
extern "C" void kernel_launch(void* const* d_in, const int* in_sizes, int n_in,
                              void* d_out, int out_size, void* d_ws, size_t ws_size,
                              hipStream_t stream) {
    const float* v_i = (const float*)d_in[0];
    const float* v_j = (const float*)d_in[1];
    const int*   adj = (const int*)d_in[2];
    const float* W   = (const float*)d_in[3];
    const float* a   = (const float*)d_in[4];
    float* out = (float*)d_out;

    float* ws  = (float*)d_ws;
    float* wa1 = ws;
    float* wa2 = ws + FIN;
    float* s_i = ws + 2 * FIN;
    float* s_j = ws + 2 * FIN + N;

    make_wa_kernel<<<1, 256, 0, stream>>>(W, a, wa1, wa2);

    dim3 grid_dot(N / 16, 2);
    rowdot_wmma_kernel<<<grid_dot, 32, 0, stream>>>(v_i, v_j, wa1, wa2, s_i, s_j);

    gat_softmax_kernel<<<N, TPB, 0, stream>>>(adj, s_i, s_j, out);
}